// MoE_21096879358054
// MI455X (gfx1250) — compile-verified
//
#include <hip/hip_runtime.h>
#include <hip/hip_bf16.h>

typedef __attribute__((ext_vector_type(2))) float v2f;
typedef __attribute__((ext_vector_type(8))) float v8f;

#define BN   32768   // B*N tokens
#define Dd   256
#define Hh   512
#define Ee   8
#define Kk   4
#define MT   64      // tokens per expert tile
#define HC   32      // H chunk width
#define NCH  (Hh / HC)
#define XP   260     // Xs row pitch (floats), bank-conflict padding
#define W1P  36      // W1s row pitch (144B, 16B aligned)
#define W2P  260     // W2s row pitch (1040B, 16B aligned)
#define HP   36      // Hs row pitch

// ---- CDNA5 async global->LDS copy (ASYNCcnt-tracked) ----------------------
__device__ __forceinline__ void async_ld_b128(unsigned lds_addr, unsigned gofs,
                                              const void* sbase) {
  asm volatile("global_load_async_to_lds_b128 %0, %1, %2"
               :: "v"(lds_addr), "v"(gofs), "s"(sbase) : "memory");
}
__device__ __forceinline__ void wait_async0() {
  asm volatile("s_wait_asynccnt 0x0" ::: "memory");
}
// low 32 bits of a flat LDS-aperture pointer == LDS byte address
__device__ __forceinline__ unsigned lds_addr_of(const void* p) {
  return (unsigned)(uintptr_t)p;
}

// ---------------------------------------------------------------- gating
__global__ void moe_gate(const float* __restrict__ x, const int* __restrict__ mask,
                         const float* __restrict__ wg, float* __restrict__ gates) {
  __shared__ float wgs[Dd * Ee];
  const int tid = threadIdx.x;
  for (int i = tid; i < Dd * Ee; i += blockDim.x) wgs[i] = wg[i];
  __syncthreads();

  const int tok = blockIdx.x * blockDim.x + tid;
  if (tok >= BN) return;

  const float4* x4 = (const float4*)(x + (size_t)tok * Dd);
  float lg[Ee];
#pragma unroll
  for (int e = 0; e < Ee; ++e) lg[e] = 0.f;
  for (int d4 = 0; d4 < Dd / 4; ++d4) {
    float4 xv = x4[d4];
    const float* w0 = &wgs[(d4 * 4) * Ee];
#pragma unroll
    for (int e = 0; e < Ee; ++e)
      lg[e] += xv.x * w0[e] + xv.y * w0[Ee + e] + xv.z * w0[2 * Ee + e] + xv.w * w0[3 * Ee + e];
  }

  float val[Kk]; int sel[Kk]; unsigned chosen = 0;
#pragma unroll
  for (int k = 0; k < Kk; ++k) {
    float best = -3.402823466e38f; int bi = 0;
#pragma unroll
    for (int e = 0; e < Ee; ++e)
      if (!(chosen & (1u << e)) && lg[e] > best) { best = lg[e]; bi = e; }
    chosen |= 1u << bi; val[k] = best; sel[k] = bi;
  }
  float p[Kk], s = 0.f;
#pragma unroll
  for (int k = 0; k < Kk; ++k) { p[k] = __expf(val[k] - val[0]); s += p[k]; }
  const float scale = (float)mask[tok] / s;

#pragma unroll
  for (int e = 0; e < Ee; ++e) {
    float g = 0.f;
#pragma unroll
    for (int k = 0; k < Kk; ++k) if (sel[k] == e) g = p[k] * scale;
    gates[(size_t)tok * Ee + e] = g;
  }
}

// ------------------------------------------------- deterministic dispatch
__global__ void moe_build(const float* __restrict__ gates, int* __restrict__ counts,
                          int* __restrict__ lists) {
  const int e = blockIdx.x, tid = threadIdx.x;
  __shared__ int sc[256];
  int running = 0;
  int* lst = lists + (size_t)e * BN;
  for (int base = 0; base < BN; base += 256) {
    const int tok = base + tid;
    const int f = (gates[(size_t)tok * Ee + e] != 0.0f) ? 1 : 0;
    sc[tid] = f;
    __syncthreads();
    for (int off = 1; off < 256; off <<= 1) {
      int v = sc[tid];
      if (tid >= off) v += sc[tid - off];
      __syncthreads();
      sc[tid] = v;
      __syncthreads();
    }
    if (f) lst[running + sc[tid] - 1] = tok;
    running += sc[255];
    __syncthreads();
  }
  if (tid == 0) counts[e] = running;
}

// ------------------------------------------------------- residual init
__global__ void moe_init(const float* __restrict__ x, float* __restrict__ out) {
  const size_t i = (size_t)blockIdx.x * blockDim.x + threadIdx.x;
  if (i < (size_t)BN * Dd / 4) ((float4*)out)[i] = ((const float4*)x)[i];
  if (i == 0) out[(size_t)BN * Dd] = 0.0f;   // moe_loss
}

// --------------------------------------------- fused expert MLP (WMMA fp32)
__launch_bounds__(256, 1)
__global__ void moe_expert(const float* __restrict__ x,
                           const float* __restrict__ W1, const float* __restrict__ b1,
                           const float* __restrict__ W2, const float* __restrict__ b2,
                           const float* __restrict__ gates,
                           const int* __restrict__ counts, const int* __restrict__ lists,
                           float* __restrict__ out) {
  extern __shared__ float lds[];
  float* Xs   = lds;                       // [MT][XP]
  float* W1sb = Xs + MT * XP;              // 2 x [Dd][W1P]
  float* W2sb = W1sb + 2 * Dd * W1P;       // 2 x [HC][W2P]
  float* Hs   = W2sb + 2 * HC * W2P;       // [MT][HP]
  float* gv   = Hs + MT * HP;              // [MT]
  int*   tks  = (int*)(gv + MT);           // [MT]

  const int e = blockIdx.x >> 9;           // / (BN/MT)
  const int t = blockIdx.x & 511;
  const int cnt = counts[e];
  const int start = t * MT;
  if (start >= cnt) return;

  const int tid  = threadIdx.x;
  const int wave = tid >> 5, lane = tid & 31;
  const int wr   = wave & 3;               // token-row group (16 rows)
  const int wc   = wave >> 2;              // 0/1
  const int hlf  = (lane < 16) ? 0 : 2;
  const int ln   = lane & 15;

  const float* W1e = W1 + (size_t)e * Dd * Hh;
  const float* W2e = W2 + (size_t)e * Hh * Dd;

  const int* lst = lists + (size_t)e * BN;
  if (tid < MT) {
    const int r = start + tid;
    const int tok = (r < cnt) ? lst[r] : 0;
    tks[tid] = tok;
    gv[tid]  = (r < cnt) ? gates[(size_t)tok * Ee + e] : 0.f;
  }
  __syncthreads();

  // gather X tile [MT x Dd] into LDS
  for (int i = tid; i < MT * (Dd / 4); i += 256) {
    const int r = i >> 6, c4 = i & 63;
    float4 xv = ((const float4*)(x + (size_t)tks[r] * Dd))[c4];
    float* dst = &Xs[r * XP + c4 * 4];
    dst[0] = xv.x; dst[1] = xv.y; dst[2] = xv.z; dst[3] = xv.w;
  }

  // kick off async weight staging for chunk 0 into buffer 0
  {
#pragma unroll
    for (int q = 0; q < 8; ++q) {                 // W1 chunk: 2048 x b128
      const int i = tid + q * 256, d = i >> 3, c4 = i & 7;
      async_ld_b128(lds_addr_of(&W1sb[d * W1P + c4 * 4]),
                    (unsigned)((d * Hh + c4 * 4) * 4), W1e);
    }
#pragma unroll
    for (int q = 0; q < 8; ++q) {                 // W2 chunk: 2048 x b128
      const int i = tid + q * 256, h = i >> 6, c4 = i & 63;
      async_ld_b128(lds_addr_of(&W2sb[h * W2P + c4 * 4]),
                    (unsigned)((h * Dd + c4 * 4) * 4), W2e);
    }
  }
  __syncthreads();   // Xs visible to all waves

  v8f yacc[8];
#pragma unroll
  for (int j = 0; j < 8; ++j)
#pragma unroll
    for (int q = 0; q < 8; ++q) yacc[j][q] = 0.f;

  for (int hc = 0; hc < NCH; ++hc) {
    const float* W1c = W1sb + (hc & 1) * Dd * W1P;
    const float* W2c = W2sb + (hc & 1) * HC * W2P;

    wait_async0();        // my async loads for current buffer done
    __syncthreads();      // everyone's done; prev reads of other buffer done

    if (hc + 1 < NCH) {   // stream next chunk into the other buffer
      float* W1n = W1sb + ((hc + 1) & 1) * Dd * W1P;
      float* W2n = W2sb + ((hc + 1) & 1) * HC * W2P;
#pragma unroll
      for (int q = 0; q < 8; ++q) {
        const int i = tid + q * 256, d = i >> 3, c4 = i & 7;
        async_ld_b128(lds_addr_of(&W1n[d * W1P + c4 * 4]),
                      (unsigned)((d * Hh + (hc + 1) * HC + c4 * 4) * 4), W1e);
      }
#pragma unroll
      for (int q = 0; q < 8; ++q) {
        const int i = tid + q * 256, h = i >> 6, c4 = i & 63;
        async_ld_b128(lds_addr_of(&W2n[h * W2P + c4 * 4]),
                      (unsigned)((((hc + 1) * HC + h) * Dd + c4 * 4) * 4), W2e);
      }
    }

    // stage 1: h tile [16x16] rows 16*wr, chunk-cols 16*wc ; K=256
    // two independent accumulation chains for WMMA ILP
    v8f h0, h1;
#pragma unroll
    for (int q = 0; q < 8; ++q) { h0[q] = 0.f; h1[q] = 0.f; }
    const float* arow = &Xs[(wr * 16 + ln) * XP + hlf];
    const float* bcol = &W1c[hlf * W1P + wc * 16 + ln];
#pragma unroll
    for (int kk = 0; kk < 64; kk += 2) {
      v2f a0; a0.x = arow[kk * 4];     a0.y = arow[kk * 4 + 1];
      v2f a1; a1.x = arow[kk * 4 + 4]; a1.y = arow[kk * 4 + 5];
      const float* b0p = bcol + (kk * 4) * W1P;
      const float* b1p = bcol + (kk * 4 + 4) * W1P;
      v2f b0; b0.x = b0p[0]; b0.y = b0p[W1P];
      v2f b1; b1.x = b1p[0]; b1.y = b1p[W1P];
      h0 = __builtin_amdgcn_wmma_f32_16x16x4_f32(false, a0, false, b0, (short)0, h0, false, false);
      h1 = __builtin_amdgcn_wmma_f32_16x16x4_f32(false, a1, false, b1, (short)0, h1, false, false);
    }
    // merge partials + bias + relu -> Hs
    const float bias = b1[e * Hh + hc * HC + wc * 16 + ln];
#pragma unroll
    for (int i2 = 0; i2 < 8; ++i2) {
      float hv = h0[i2] + h1[i2] + bias;
      hv = hv > 0.f ? hv : 0.f;
      const int row = wr * 16 + i2 + ((lane < 16) ? 0 : 8);
      Hs[row * HP + wc * 16 + ln] = hv;
    }
    __syncthreads();

    // stage 2: Y[16 x 128] rows 16*wr, cols 128*wc ; K=HC, 8 indep chains
    const float* hrow = &Hs[(wr * 16 + ln) * HP + hlf];
#pragma unroll
    for (int k = 0; k < HC / 4; ++k) {
      v2f a; a.x = hrow[k * 4]; a.y = hrow[k * 4 + 1];
      v2f bb[8];
#pragma unroll
      for (int j = 0; j < 8; ++j) {
        const float* bp = &W2c[(k * 4 + hlf) * W2P + wc * 128 + j * 16 + ln];
        bb[j].x = bp[0]; bb[j].y = bp[W2P];
      }
#pragma unroll
      for (int j = 0; j < 8; ++j)
        yacc[j] = __builtin_amdgcn_wmma_f32_16x16x4_f32(false, a, false, bb[j],
                                                        (short)0, yacc[j], false, false);
    }
  }

  // epilogue: (Y + b2) * gate scattered into out with f32 atomics
#pragma unroll
  for (int j = 0; j < 8; ++j) {
    const int col = wc * 128 + j * 16 + ln;
    const float bb = b2[e * Dd + col];
#pragma unroll
    for (int i2 = 0; i2 < 8; ++i2) {
      const int row = wr * 16 + i2 + ((lane < 16) ? 0 : 8);
      const float g = gv[row];
      if (g != 0.f) {
        const float v = (yacc[j][i2] + bb) * g;
        unsafeAtomicAdd(&out[(size_t)tks[row] * Dd + col], v);
      }
    }
  }
}

// ---------------------------------------------------------------- launch
extern "C" void kernel_launch(void* const* d_in, const int* in_sizes, int n_in,
                              void* d_out, int out_size, void* d_ws, size_t ws_size,
                              hipStream_t stream) {
  const float* x    = (const float*)d_in[0];
  const int*   mask = (const int*)d_in[1];
  const float* wg   = (const float*)d_in[2];
  const float* W1   = (const float*)d_in[3];
  const float* b1   = (const float*)d_in[4];
  const float* W2   = (const float*)d_in[5];
  const float* b2   = (const float*)d_in[6];
  float* out = (float*)d_out;

  float* gates  = (float*)d_ws;                                    // BN*Ee f32
  int*   counts = (int*)((char*)d_ws + (size_t)BN * Ee * 4);       // Ee ints
  int*   lists  = (int*)((char*)d_ws + (size_t)BN * Ee * 4 + 256); // Ee*BN ints

  constexpr size_t LDSB =
      (size_t)(MT * XP + 2 * Dd * W1P + 2 * HC * W2P + MT * HP + MT) * sizeof(float) +
      (size_t)MT * sizeof(int);   // ~212 KB: fits 320 KB/WGP LDS on CDNA5

  moe_gate  <<<BN / 256, 256, 0, stream>>>(x, mask, wg, gates);
  moe_build <<<Ee, 256, 0, stream>>>(gates, counts, lists);
  moe_init  <<<(BN * Dd / 4 + 255) / 256, 256, 0, stream>>>(x, out);
  moe_expert<<<Ee * (BN / MT), 256, LDSB, stream>>>(x, W1, b1, W2, b2,
                                                    gates, counts, lists, out);
}